// OtsuThreshold_1382979469334
// MI455X (gfx1250) — compile-verified
//
#include <hip/hip_runtime.h>

#define NB 256
#define ROW_SHIFT 20
#define ROW_ELEMS (1u << ROW_SHIFT)   // 1024*1024 elements per (b*c) row

typedef float v2f __attribute__((ext_vector_type(2)));
typedef float v8f __attribute__((ext_vector_type(8)));

// ---- order-preserving float <-> uint key (works for any sign) ----
__device__ __forceinline__ unsigned flip_f32(float f) {
    unsigned u = __float_as_uint(f);
    return u ^ ((unsigned)((int)u >> 31) | 0x80000000u);
}
__device__ __forceinline__ float unflip_f32(unsigned k) {
    unsigned u = (k & 0x80000000u) ? (k ^ 0x80000000u) : ~k;
    return __uint_as_float(u);
}

// ---------------- pass 0: init workspace ----------------
__global__ void otsu_init_kernel(unsigned* __restrict__ minmax,
                                 unsigned* __restrict__ hist,
                                 float* __restrict__ thr, int nrows) {
    int tid = threadIdx.x;
    if (tid == 0) { minmax[0] = 0xFFFFFFFFu; minmax[1] = 0u; }
    int total = nrows * NB;
    for (int i = tid; i < total; i += blockDim.x) hist[i] = 0u;
    for (int i = tid; i < nrows; i += blockDim.x) thr[i] = 0.0f;
}

// ---------------- pass 1: global min/max ----------------
__global__ void otsu_minmax_kernel(const float4* __restrict__ x4, size_t n4,
                                   unsigned* __restrict__ minmax) {
    __shared__ unsigned smn[256];
    __shared__ unsigned smx[256];
    unsigned lmn = 0xFFFFFFFFu, lmx = 0u;
    size_t stride = (size_t)gridDim.x * blockDim.x;
    for (size_t i = (size_t)blockIdx.x * blockDim.x + threadIdx.x; i < n4; i += stride) {
        __builtin_prefetch(&x4[i + stride], 0, 0);
        float4 v = x4[i];
        unsigned a = flip_f32(v.x), b = flip_f32(v.y);
        unsigned c = flip_f32(v.z), d = flip_f32(v.w);
        lmn = min(lmn, min(min(a, b), min(c, d)));
        lmx = max(lmx, max(max(a, b), max(c, d)));
    }
    smn[threadIdx.x] = lmn;
    smx[threadIdx.x] = lmx;
    __syncthreads();
    for (int off = 128; off > 0; off >>= 1) {
        if ((int)threadIdx.x < off) {
            smn[threadIdx.x] = min(smn[threadIdx.x], smn[threadIdx.x + off]);
            smx[threadIdx.x] = max(smx[threadIdx.x], smx[threadIdx.x + off]);
        }
        __syncthreads();
    }
    if (threadIdx.x == 0) {
        atomicMin(&minmax[0], smn[0]);
        atomicMax(&minmax[1], smx[0]);
    }
}

// ---------------- pass 2: per-row histograms ----------------
// 256 threads = 8 wave32s; per-wave private LDS histograms (ds_add_u32),
// then block-reduce and one global atomicAdd per bin.
__global__ void otsu_hist_kernel(const float* __restrict__ x,
                                 unsigned* __restrict__ hist,
                                 const unsigned* __restrict__ minmax,
                                 int blocksPerRow) {
    __shared__ unsigned lh[8][NB];
    int wv = threadIdx.x >> 5;
    int row = blockIdx.x / blocksPerRow;
    int chunk = blockIdx.x - row * blocksPerRow;
#pragma unroll
    for (int w = 0; w < 8; ++w) lh[w][threadIdx.x] = 0u;
    __syncthreads();

    float mn = unflip_f32(minmax[0]);
    float mx = unflip_f32(minmax[1]);
    float invw = (float)NB / (mx - mn);   // 1/width, width = (mx-mn)/nbins

    const float4* xr = (const float4*)(x + (size_t)row * ROW_ELEMS);
    int per  = (int)(ROW_ELEMS / 4) / blocksPerRow;  // float4 per block
    int base = chunk * per;
    for (int i = threadIdx.x; i < per; i += blockDim.x) {
        float4 v = xr[base + i];
        int b0 = min(max((int)floorf((v.x - mn) * invw), 0), NB - 1);
        int b1 = min(max((int)floorf((v.y - mn) * invw), 0), NB - 1);
        int b2 = min(max((int)floorf((v.z - mn) * invw), 0), NB - 1);
        int b3 = min(max((int)floorf((v.w - mn) * invw), 0), NB - 1);
        atomicAdd(&lh[wv][b0], 1u);
        atomicAdd(&lh[wv][b1], 1u);
        atomicAdd(&lh[wv][b2], 1u);
        atomicAdd(&lh[wv][b3], 1u);
    }
    __syncthreads();
    unsigned s = 0;
#pragma unroll
    for (int w = 0; w < 8; ++w) s += lh[w][threadIdx.x];
    if (s) atomicAdd(&hist[(size_t)row * NB + threadIdx.x], s);
}

// ---------------- pass 3: Otsu via WMMA triangular matmul ----------------
// One wave per 16-row tile. Cumulative sums CW = H @ U, CS = (k*H) @ U with
// U upper-triangular ones, computed with V_WMMA_F32_16X16X4_F32. Counts are
// <= 2^20 so all sums are exact in f32 accumulators.
__global__ __launch_bounds__(32) void otsu_scan_wmma_kernel(
        const unsigned* __restrict__ hist, const unsigned* __restrict__ minmax,
        float* __restrict__ ws_thr, float* __restrict__ out_thr) {
    __shared__ float lh [16 * NB];
    __shared__ float lcw[16 * NB];
    __shared__ float lcs[16 * NB];

    const int lane  = threadIdx.x;        // 0..31, all active (EXEC all-1s)
    const int hv    = lane >> 4;          // lane half (0/1)
    const int m     = lane & 15;          // M (A) / N (B,C,D) index
    const int rbase = blockIdx.x * 16;

    // stage 16 rows x 256 bins of counts into LDS as f32
    for (int i = lane; i < 16 * NB; i += 32)
        lh[i] = (float)hist[(size_t)rbase * NB + i];
    __syncthreads();

    // column tiles jt: CW/CS[:, 16*jt .. 16*jt+15]
    for (int jt = 0; jt < 16; ++jt) {
        v8f cw = {0, 0, 0, 0, 0, 0, 0, 0};
        v8f cs = {0, 0, 0, 0, 0, 0, 0, 0};
        for (int kt = 0; kt <= jt; ++kt) {
            const bool diag = (kt == jt);
#pragma unroll
            for (int ks = 0; ks < 4; ++ks) {
                // A 16x4 layout: lane holds M=m, K = k0, k0+1
                const int k0 = kt * 16 + ks * 4 + hv * 2;
                v2f a, as, b;
                a.x  = lh[m * NB + k0];
                a.y  = lh[m * NB + k0 + 1];
                as.x = a.x * (float)k0;
                as.y = a.y * (float)(k0 + 1);
                // B 4x16 layout: lane holds N=m, rows K = 2*hv, 2*hv+1
                if (diag) {
                    const int kk = ks * 4 + hv * 2;   // k within 16-block
                    b.x = (kk     <= m) ? 1.0f : 0.0f;
                    b.y = (kk + 1 <= m) ? 1.0f : 0.0f;
                } else {
                    b.x = 1.0f;
                    b.y = 1.0f;
                }
                cw = __builtin_amdgcn_wmma_f32_16x16x4_f32(
                        false, a, false, b, (short)0, cw, false, false);
                cs = __builtin_amdgcn_wmma_f32_16x16x4_f32(
                        false, as, false, b, (short)0, cs, false, false);
            }
        }
        // D 16x16 layout: VGPR v -> row v + 8*hv, col = m
#pragma unroll
        for (int v = 0; v < 8; ++v) {
            const int rl  = v + hv * 8;
            const int col = jt * 16 + m;
            lcw[rl * NB + col] = cw[v];
            lcs[rl * NB + col] = cs[v];
        }
    }
    __syncthreads();

    const float mn = unflip_f32(minmax[0]);
    const float mx = unflip_f32(minmax[1]);
    const float edge = (mx - mn) / (float)(NB - 1);   // bin_edges spacing

    for (int rl = 0; rl < 16; ++rl) {
        const float totw = lcw[rl * NB + (NB - 1)];
        const float tots = lcs[rl * NB + (NB - 1)];
        const float invN = 1.0f / totw;               // normalize like reference
        const float tsum = tots * invN;
        float bestv = -2.0f;
        int   bestt = 0;
        for (int t = lane; t < NB - 1; t += 32) {
            float wb = lcw[rl * NB + t] * invN;
            float sb = lcs[rl * NB + t] * invN;
            float wf = 1.0f - wb;
            float sf = tsum - sb;
            float mb = (wb > 0.0f) ? sb / wb : 0.0f;
            float mf = (wf > 0.0f) ? sf / wf : 0.0f;
            float d  = mb - mf;
            float var = ((wb > 0.0f) && (wf > 0.0f)) ? wb * wf * d * d : -1.0f;
            if (var > bestv) { bestv = var; bestt = t; }
        }
        // wave argmax; ties -> lowest t (matches jnp.argmax)
        for (int off = 16; off > 0; off >>= 1) {
            float ov = __shfl_xor(bestv, off, 32);
            int   ot = __shfl_xor(bestt, off, 32);
            if (ov > bestv || (ov == bestv && ot < bestt)) { bestv = ov; bestt = ot; }
        }
        if (lane == 0) {
            const int row = rbase + rl;
            const float thr = (bestv > 0.0f) ? (mn + (float)(bestt + 1) * edge) : 0.0f;
            ws_thr[row]  = thr;
            out_thr[row] = thr;
        }
    }
}

// ---------------- pass 4: apply threshold ----------------
__global__ void otsu_apply_kernel(const float4* __restrict__ x4,
                                  float4* __restrict__ o4,
                                  const float* __restrict__ thr, size_t n4) {
    size_t stride = (size_t)gridDim.x * blockDim.x;
    for (size_t i = (size_t)blockIdx.x * blockDim.x + threadIdx.x; i < n4; i += stride) {
        __builtin_prefetch(&x4[i + stride], 0, 0);
        const int row = (int)((i * 4) >> ROW_SHIFT);
        const float t = thr[row];
        float4 v = x4[i];
        float4 r;
        r.x = (v.x > t) ? v.x : 0.0f;
        r.y = (v.y > t) ? v.y : 0.0f;
        r.z = (v.z > t) ? v.z : 0.0f;
        r.w = (v.w > t) ? v.w : 0.0f;
        o4[i] = r;
    }
}

extern "C" void kernel_launch(void* const* d_in, const int* in_sizes, int n_in,
                              void* d_out, int out_size, void* d_ws, size_t ws_size,
                              hipStream_t stream) {
    (void)n_in; (void)out_size; (void)ws_size;
    const float* x = (const float*)d_in[0];
    const size_t N = (size_t)in_sizes[0];          // 16*3*1024*1024
    const int nrows = (int)(N >> ROW_SHIFT);       // 48
    float* out = (float*)d_out;

    // workspace layout: [0..1] minmax keys | hist[nrows][256] | thr[nrows]
    unsigned* minmax = (unsigned*)d_ws;
    unsigned* hist   = (unsigned*)((char*)d_ws + 16);
    float*    thr    = (float*)((char*)d_ws + 16 + (size_t)nrows * NB * sizeof(unsigned));

    otsu_init_kernel<<<1, 256, 0, stream>>>(minmax, hist, thr, nrows);

    otsu_minmax_kernel<<<4096, 256, 0, stream>>>((const float4*)x, N / 4, minmax);

    const int blocksPerRow = 32;                   // 32768 elems per block
    otsu_hist_kernel<<<nrows * blocksPerRow, 256, 0, stream>>>(x, hist, minmax, blocksPerRow);

    otsu_scan_wmma_kernel<<<nrows / 16, 32, 0, stream>>>(hist, minmax, thr, out + N);

    otsu_apply_kernel<<<8192, 256, 0, stream>>>((const float4*)x, (float4*)out, thr, N / 4);
}